// Model_13572096655821
// MI455X (gfx1250) — compile-verified
//
#include <hip/hip_runtime.h>

// Per-channel histogram: array [1e6 x 64] int32 in [0,256) -> counts [64 x 256] int32.
// Memory-bound (256 MB read @ 23.3 TB/s ~= 11 us floor). Strategy:
//   - async global->LDS staging pipeline (4-deep, ASYNCcnt-managed)
//   - LDS-privatized per-block histogram with ds atomics (padded to kill bank conflicts)
//   - packed u64 global-atomic flush (2 bins per atomic, L2-resident output)

#define TPB           256
#define CHANNELS      64
#define BINS          256
#define HSTRIDE       257                 // pad: LDS bank = (c + bin) & 63
#define ROWS_PER_TILE 32                  // 32 rows * 64 ch * 4B = 8 KB per stage
#define NBUF          4
#define STAGE_INTS    (2 * TPB * 4)       // 2048 ints = 8 KB per buffer
#define GRID          320                 // keep flush-atomic count bounded

__device__ __forceinline__ void wait_async_stages(int inflight) {
    // Each stage issues 2 async ops per wave; ops complete in order.
    // Waiting to (2*inflight - 2) guarantees the oldest stage has fully landed in LDS.
    if (inflight >= 4)      asm volatile("s_wait_asynccnt 6" ::: "memory");
    else if (inflight == 3) asm volatile("s_wait_asynccnt 4" ::: "memory");
    else if (inflight == 2) asm volatile("s_wait_asynccnt 2" ::: "memory");
    else                    asm volatile("s_wait_asynccnt 0" ::: "memory");
}

__global__ __launch_bounds__(TPB) void hist_zero(int* __restrict__ out, int n) {
    int i = blockIdx.x * blockDim.x + threadIdx.x;
    if (i < n) out[i] = 0;
}

__global__ __launch_bounds__(TPB) void hist_main(const int* __restrict__ arr,
                                                 unsigned long long* __restrict__ out,
                                                 int nRows) {
    __shared__ unsigned int hist[CHANNELS * HSTRIDE];           // 65792 B
    __shared__ __align__(16) int stage[NBUF * STAGE_INTS];      // 32768 B

    const int tid = threadIdx.x;
    for (int i = tid; i < CHANNELS * HSTRIDE; i += TPB) hist[i] = 0u;
    __syncthreads();

    const int laneRow = tid >> 4;          // 0..15 : row within half-tile
    const int chan4   = (tid & 15) << 2;   // 0,4,...,60 : 4 consecutive channels
    const int nTiles  = nRows / ROWS_PER_TILE;
    const int stride  = gridDim.x;

    // Low 32 bits of a generic LDS pointer == LDS byte offset (aperture mapping).
    const unsigned ldsBase = (unsigned)(size_t)&stage[0] + (unsigned)tid * 16u;

    auto issue = [&](int b, int tile) {
        const int* p0 = arr + ((size_t)tile * ROWS_PER_TILE + (size_t)laneRow) * CHANNELS + chan4;
        unsigned l0 = ldsBase + (unsigned)(b * STAGE_INTS * 4);
        asm volatile("global_load_async_to_lds_b128 %0, %1, off"
                     :: "v"(l0), "v"((unsigned long long)(size_t)p0) : "memory");
        asm volatile("global_load_async_to_lds_b128 %0, %1, off"
                     :: "v"((unsigned)(l0 + TPB * 16u)),
                        "v"((unsigned long long)(size_t)(p0 + 16 * CHANNELS)) : "memory");
    };

    int inflight = 0;
    int nextTile = blockIdx.x;
    for (int b = 0; b < NBUF && nextTile < nTiles; ++b) {
        issue(b, nextTile);
        nextTile += stride;
        ++inflight;
    }

    int buf = 0;
    for (int t = blockIdx.x; t < nTiles; t += stride) {
        wait_async_stages(inflight);

        // Each wave reads back exactly the LDS slots it staged -> no barrier needed.
        const int4 a = *(const int4*)&stage[buf * STAGE_INTS + tid * 4];
        const int4 c = *(const int4*)&stage[buf * STAGE_INTS + TPB * 4 + tid * 4];

        atomicAdd(&hist[(chan4 + 0) * HSTRIDE + (a.x & 255)], 1u);
        atomicAdd(&hist[(chan4 + 1) * HSTRIDE + (a.y & 255)], 1u);
        atomicAdd(&hist[(chan4 + 2) * HSTRIDE + (a.z & 255)], 1u);
        atomicAdd(&hist[(chan4 + 3) * HSTRIDE + (a.w & 255)], 1u);
        atomicAdd(&hist[(chan4 + 0) * HSTRIDE + (c.x & 255)], 1u);
        atomicAdd(&hist[(chan4 + 1) * HSTRIDE + (c.y & 255)], 1u);
        atomicAdd(&hist[(chan4 + 2) * HSTRIDE + (c.z & 255)], 1u);
        atomicAdd(&hist[(chan4 + 3) * HSTRIDE + (c.w & 255)], 1u);

        --inflight;
        if (nextTile < nTiles) {          // refill this buffer for a future tile
            issue(buf, nextTile);
            nextTile += stride;
            ++inflight;
        }
        buf = (buf + 1) & (NBUF - 1);
    }

    // Tail rows (nRows not a multiple of 32): block 0, direct loads.
    const int rowsMain = nTiles * ROWS_PER_TILE;
    if (blockIdx.x == 0 && rowsMain < nRows) {
        const int tailElems = (nRows - rowsMain) * CHANNELS;
        const int* tp = arr + (size_t)rowsMain * CHANNELS;
        for (int i = tid; i < tailElems; i += TPB) {
            atomicAdd(&hist[(i & 63) * HSTRIDE + (tp[i] & 255)], 1u);
        }
    }
    __syncthreads();

    // Flush: pack two adjacent 32-bit bins into one u64 atomic.
    // Per-(channel,bin) totals <= nRows < 2^31, so the low word never carries.
    for (int idx = tid; idx < CHANNELS * (BINS / 2); idx += TPB) {
        const int c  = idx >> 7;           // / 128
        const int b0 = (idx & 127) << 1;
        unsigned long long lo = hist[c * HSTRIDE + b0];
        unsigned long long hi = hist[c * HSTRIDE + b0 + 1];
        unsigned long long v  = lo | (hi << 32);
        if (v) atomicAdd(&out[idx], v);    // u64 slot idx == int32 slots [c*256+b0, +1]
    }
}

extern "C" void kernel_launch(void* const* d_in, const int* in_sizes, int n_in,
                              void* d_out, int out_size, void* d_ws, size_t ws_size,
                              hipStream_t stream) {
    (void)n_in; (void)d_ws; (void)ws_size;
    const int* arr = (const int*)d_in[0];
    const int nRows = in_sizes[0] / CHANNELS;

    hist_zero<<<(out_size + TPB - 1) / TPB, TPB, 0, stream>>>((int*)d_out, out_size);
    hist_main<<<GRID, TPB, 0, stream>>>(arr, (unsigned long long*)d_out, nRows);
}